// AttnBlock_3332894621889
// MI455X (gfx1250) — compile-verified
//
#include <hip/hip_runtime.h>

#define CH 128      // channels
#define NG 8        // groupnorm groups
#define HH 4        // heads
#define DD 32       // per-head dim

typedef __attribute__((ext_vector_type(2))) float v2f;
typedef __attribute__((ext_vector_type(8))) float v8f;

__device__ __forceinline__ unsigned f2ord(float f) {
  unsigned u = __float_as_uint(f);
  return (u & 0x80000000u) ? ~u : (u | 0x80000000u);
}
__device__ __forceinline__ float ord2f(unsigned u) {
  return __uint_as_float((u & 0x80000000u) ? (u & 0x7FFFFFFFu) : ~u);
}

// ---------------------------------------------------------------- max(ew)
__global__ void k_init_max(unsigned* mx) { *mx = 0u; }

__global__ __launch_bounds__(256) void k_maxew(const float* __restrict__ ew,
                                               unsigned* __restrict__ mx, int E) {
  int t = blockIdx.x * blockDim.x + threadIdx.x;
  if (t < E) atomicMax(mx, f2ord(ew[t]));
}

// ------------------------------------------------- edge coefficient per head
__global__ void k_coef(const float* __restrict__ We, const float* __restrict__ a_edge,
                       float* __restrict__ coef) {
  int h = threadIdx.x;
  if (h < HH) {
    float s = 0.f;
    for (int d = 0; d < DD; ++d) s += We[h * DD + d] * a_edge[h * DD + d];
    coef[h] = s;
  }
}

// --------------------------------------------- GroupNorm + LeakyReLU(0.01)
__global__ __launch_bounds__(256) void k_gnorm_act(const float* __restrict__ x,
                                                   const float* __restrict__ gamma,
                                                   const float* __restrict__ beta,
                                                   float* __restrict__ y, int total) {
  int t = blockIdx.x * blockDim.x + threadIdx.x;   // node*NG + g
  if (t >= total) return;
  int node = t >> 3, g = t & 7;
  const float* xp = x + (size_t)node * CH + g * 16;
  float v[16], mu = 0.f, m2 = 0.f;
#pragma unroll
  for (int i = 0; i < 16; ++i) { v[i] = xp[i]; mu += v[i]; }
  mu *= (1.f / 16.f);
#pragma unroll
  for (int i = 0; i < 16; ++i) { float d = v[i] - mu; m2 += d * d; }
  float rs = rsqrtf(m2 * (1.f / 16.f) + 1e-5f);
  const float* gp = gamma + g * 16;
  const float* bp = beta + g * 16;
  float* yp = y + (size_t)node * CH + g * 16;
#pragma unroll
  for (int i = 0; i < 16; ++i) {
    float xn = (v[i] - mu) * rs * gp[i] + bp[i];
    yp[i] = xn > 0.f ? xn : 0.01f * xn;
  }
}

// ----------------------------------------------------- h = xn @ W  (WMMA f32)
// Weights staged once per block into LDS in fragment-friendly layout:
//   lds[((k>>2)*CH + n)*4 + (k&3)] = W[k*CH + n]
// so a lane's B fragment (two K-adjacent values of its column) is one b64 LDS read.
// Per k-step: load A (b64 global) + all 8 B fragments (4x ds_load_2addr_b64,
// single dscnt wait), then 8 back-to-back wmmas.
__global__ __launch_bounds__(256) void k_gemm(const float* __restrict__ A,
                                              const float* __restrict__ B,
                                              float* __restrict__ C, int M) {
  extern __shared__ float lds[];   // CH*CH floats = 64 KB
  for (int idx = threadIdx.x; idx < CH * CH; idx += blockDim.x) {
    int k = idx >> 7, n = idx & (CH - 1);
    lds[(((k >> 2) * CH) + n) * 4 + (k & 3)] = B[idx];   // coalesced global read
  }
  __syncthreads();

  int wave = blockIdx.x * (blockDim.x >> 5) + (threadIdx.x >> 5);
  int m0 = wave * 16;
  if (m0 >= M) return;                    // whole-wave uniform exit, EXEC stays full
  int lane = threadIdx.x & 31;
  int row = lane & 15;                    // M (A) / N (B) index within tile
  int hi  = lane >> 4;                    // half-wave -> K-pair select
  int r = m0 + row; if (r >= M) r = M - 1;
  const float* arow = A + (size_t)r * CH;

  v8f acc[8];
#pragma unroll
  for (int t = 0; t < 8; ++t) acc[t] = (v8f){0.f,0.f,0.f,0.f,0.f,0.f,0.f,0.f};

  for (int k4 = 0; k4 < CH / 4; ++k4) {
    // A fragment: 16x4 f32: lanes 0-15 hold K=4k4+0..1, lanes 16-31 K=4k4+2..3
    v2f a = *(const v2f*)(arow + k4 * 4 + hi * 2);
    const float* bp = lds + ((size_t)k4 * CH + row) * 4 + hi * 2;
    v2f b[8];
#pragma unroll
    for (int t = 0; t < 8; ++t) b[t] = *(const v2f*)(bp + t * 64);  // clause, 1 wait
#pragma unroll
    for (int t = 0; t < 8; ++t) {
      acc[t] = __builtin_amdgcn_wmma_f32_16x16x4_f32(
          /*neg_a=*/false, a, /*neg_b=*/false, b[t],
          /*c_mod=*/(short)0, acc[t], /*reuse_a=*/false, /*reuse_b=*/false);
    }
  }
  // C/D layout: VGPR v, lane L -> (M = v + 8*(L>>4), N = L&15)
#pragma unroll
  for (int t = 0; t < 8; ++t) {
    int n0 = t * 16;
#pragma unroll
    for (int v = 0; v < 8; ++v) {
      int mrow = m0 + v + 8 * hi;
      if (mrow < M) C[(size_t)mrow * CH + n0 + row] = acc[t][v];
    }
  }
}

// ------------------------------------------- per-node attention scores (h·a)
__global__ __launch_bounds__(256) void k_scores(const float* __restrict__ h,
                                                const float* __restrict__ a_src,
                                                const float* __restrict__ a_dst,
                                                float* __restrict__ ssrc,
                                                float* __restrict__ sdst, int total) {
  int t = blockIdx.x * blockDim.x + threadIdx.x;   // node*HH + hd
  if (t >= total) return;
  int node = t >> 2, hd = t & 3;
  const float* hp = h + (size_t)node * CH + hd * DD;
  float s1 = 0.f, s2 = 0.f;
#pragma unroll 8
  for (int d = 0; d < DD; ++d) {
    float v = hp[d];
    s1 += v * a_src[hd * DD + d];
    s2 += v * a_dst[hd * DD + d];
  }
  ssrc[t] = s1;
  sdst[t] = s2;
}

// ------------------- init out=bias, m=-inf(encoded 0), ssum=0  (N*CH threads)
__global__ __launch_bounds__(256) void k_fill(const float* __restrict__ bias,
                                              float* __restrict__ out,
                                              unsigned* __restrict__ m,
                                              float* __restrict__ ssum, int total) {
  int t = blockIdx.x * blockDim.x + threadIdx.x;
  if (t >= total) return;
  int c = t & (CH - 1);
  out[t] = bias[c];
  if (c < HH) {
    int node = t >> 7;
    m[node * HH + c] = 0u;     // below encode of any real float
    ssum[node * HH + c] = 0.f;
  }
}

// --------------------------- edge pass 1: logits + leaky(0.2) + segment max
__global__ __launch_bounds__(256) void k_edge_logits(const int* __restrict__ ei,
                                                     const float* __restrict__ ew,
                                                     const float* __restrict__ ssrc,
                                                     const float* __restrict__ sdst,
                                                     const float* __restrict__ coef,
                                                     const unsigned* __restrict__ mxew,
                                                     float* __restrict__ logits,
                                                     unsigned* __restrict__ m,
                                                     int E, int total4) {
  int t = blockIdx.x * blockDim.x + threadIdx.x;   // e*HH + hd
  if (t >= total4) return;
  int e = t >> 2, hd = t & 3;
  int src, dst; float w;
  if (e < E) { src = ei[e]; dst = ei[E + e]; w = ew[e]; }
  else       { src = dst = e - E; w = ord2f(*mxew); }   // self loop, fill='max'
  float lg = ssrc[src * HH + hd] + sdst[dst * HH + hd] + w * coef[hd];
  lg = lg > 0.f ? lg : 0.2f * lg;
  logits[t] = lg;
  atomicMax(m + dst * HH + hd, f2ord(lg));
}

// --------------------------- edge pass 2: e = exp(logit - m[dst]), segment sum
__global__ __launch_bounds__(256) void k_edge_exp(const int* __restrict__ ei,
                                                  const unsigned* __restrict__ m,
                                                  float* __restrict__ logits,
                                                  float* __restrict__ ssum,
                                                  int E, int total4) {
  int t = blockIdx.x * blockDim.x + threadIdx.x;
  if (t >= total4) return;
  int e = t >> 2, hd = t & 3;
  int dst = (e < E) ? ei[E + e] : (e - E);
  float ex = expf(logits[t] - ord2f(m[dst * HH + hd]));
  logits[t] = ex;                              // reuse slot for exp value
  atomicAdd(ssum + dst * HH + hd, ex);
}

// --------------------------- edge pass 3: out[dst] += alpha * h[src] (1 wave/edge)
__global__ __launch_bounds__(256) void k_edge_aggr(const int* __restrict__ ei,
                                                   const float* __restrict__ h,
                                                   const float* __restrict__ expv,
                                                   const float* __restrict__ ssum,
                                                   float* __restrict__ out,
                                                   int E, int totalE) {
  int e = blockIdx.x * (blockDim.x >> 5) + (threadIdx.x >> 5);
  if (e >= totalE) return;
  int lane = threadIdx.x & 31;
  int src, dst;
  if (e < E) { src = ei[e]; dst = ei[E + e]; } else { src = dst = e - E; }
  int hd = lane >> 3;                           // 4 channels per lane -> head = lane/8
  float alpha = expv[e * HH + hd] / (ssum[dst * HH + hd] + 1e-16f);
  const float4 hv = *(const float4*)(h + (size_t)src * CH + lane * 4);
  float* op = out + (size_t)dst * CH + lane * 4;
  atomicAdd(op + 0, alpha * hv.x);
  atomicAdd(op + 1, alpha * hv.y);
  atomicAdd(op + 2, alpha * hv.z);
  atomicAdd(op + 3, alpha * hv.w);
}

static inline int cdiv(int a, int b) { return (a + b - 1) / b; }

extern "C" void kernel_launch(void* const* d_in, const int* in_sizes, int n_in,
                              void* d_out, int out_size, void* d_ws, size_t ws_size,
                              hipStream_t stream) {
  const int N = in_sizes[0] / CH;
  const int E = in_sizes[2];
  const int Etot = E + N;

  const float* x  = (const float*)d_in[0];
  const int*   ei = (const int*)d_in[1];
  const float* ew = (const float*)d_in[2];
  // per layer: gamma, beta, W, We, a_src, a_dst, a_edge, bias (8 entries)
  auto P = [&](int l, int i) { return (const float*)d_in[3 + l * 8 + i]; };

  float* ws = (float*)d_ws;
  size_t offA    = 0;                              // N*CH (xn; reused for logits/exp)
  size_t offH    = offA    + (size_t)N * CH;       // N*CH
  size_t offSsrc = offH    + (size_t)N * CH;       // N*HH
  size_t offSdst = offSsrc + (size_t)N * HH;
  size_t offM    = offSdst + (size_t)N * HH;       // unsigned, N*HH
  size_t offSum  = offM    + (size_t)N * HH;
  size_t offCoef = offSum  + (size_t)N * HH;       // 4
  size_t offMax  = offCoef + 4;                    // 1

  float*    bufA = ws + offA;
  float*    bufH = ws + offH;
  float*    ssrc = ws + offSsrc;
  float*    sdst = ws + offSdst;
  unsigned* mbuf = (unsigned*)(ws + offM);
  float*    ssum = ws + offSum;
  float*    coef = ws + offCoef;
  unsigned* mxew = (unsigned*)(ws + offMax);

  float* out = (float*)d_out;

  k_init_max<<<1, 1, 0, stream>>>(mxew);
  k_maxew<<<cdiv(E, 256), 256, 0, stream>>>(ew, mxew, E);

  const float* xin = x;
  for (int l = 0; l < 2; ++l) {
    const float* gamma  = P(l, 0);
    const float* beta   = P(l, 1);
    const float* W      = P(l, 2);
    const float* We     = P(l, 3);
    const float* a_src  = P(l, 4);
    const float* a_dst  = P(l, 5);
    const float* a_edge = P(l, 6);
    const float* bias   = P(l, 7);

    k_coef<<<1, 32, 0, stream>>>(We, a_edge, coef);
    k_gnorm_act<<<cdiv(N * NG, 256), 256, 0, stream>>>(xin, gamma, beta, bufA, N * NG);
    k_gemm<<<cdiv(cdiv(N, 16), 8), 256, CH * CH * sizeof(float), stream>>>(bufA, W, bufH, N);
    k_scores<<<cdiv(N * HH, 256), 256, 0, stream>>>(bufH, a_src, a_dst, ssrc, sdst, N * HH);
    k_fill<<<cdiv(N * CH, 256), 256, 0, stream>>>(bias, out, mbuf, ssum, N * CH);
    // bufA (xn) is dead after k_gemm -> reuse as logits/exp buffer (Etot*HH <= N*CH)
    k_edge_logits<<<cdiv(Etot * HH, 256), 256, 0, stream>>>(ei, ew, ssrc, sdst, coef,
                                                            mxew, bufA, mbuf, E, Etot * HH);
    k_edge_exp<<<cdiv(Etot * HH, 256), 256, 0, stream>>>(ei, mbuf, bufA, ssum, E, Etot * HH);
    k_edge_aggr<<<cdiv(Etot, 8), 256, 0, stream>>>(ei, bufH, bufA, ssum, out, E, Etot);

    xin = out;  // layer-1 GroupNorm reads layer-0 output before k_fill overwrites it
  }
}